// SoftDTW_57320633532621
// MI455X (gfx1250) — compile-verified
//
#include <hip/hip_runtime.h>
#include <hip/hip_bf16.h>
#include <math.h>

// Problem constants (from reference setup_inputs)
#define BATCH 128
#define TLEN  512   // T1 == T2 == N == M
#define CDIM  64
#define INF_V 100000000.0f

typedef float v2f __attribute__((ext_vector_type(2)));
typedef float v8f __attribute__((ext_vector_type(8)));

// ---------------------------------------------------------------------------
// Kernel 1: batched XY[b,i,j] = dot(x[b,i,:], y[b,j,:]) via V_WMMA_F32_16X16X4_F32
// One 16x16 output tile per wave32; 16 WMMAs (K=4 each) cover K=64.
//
// f32 16x16x4 VGPR layouts (ISA 7.12.2):
//   A (16x4, MxK): lane L holds row M=L&15; VGPR0/1 = K = 2*(L>>4), 2*(L>>4)+1
//   B (4x16, KxN): lane L holds col N=L&15; VGPR0/1 = K = 2*(L>>4), 2*(L>>4)+1
//   C/D (16x16):   vgpr r, lane L -> (M = r + 8*(L>>4), N = L&15)
// Since B = y^T, the per-lane gather from y is the same pattern as from x.
// ---------------------------------------------------------------------------
__global__ __launch_bounds__(256) void
softdtw_xy_wmma_kernel(const float* __restrict__ x,
                       const float* __restrict__ y,
                       float* __restrict__ xy)
{
    const int wave = threadIdx.x >> 5;
    const int lane = threadIdx.x & 31;

    // 8 tiles per block; tiles per batch = (512/16)^2 = 1024
    const int tileId = blockIdx.x * 8 + wave;
    const int b  = tileId >> 10;
    const int t  = tileId & 1023;
    const int i0 = (t >> 5) << 4;   // tile row origin
    const int j0 = (t & 31) << 4;   // tile col origin

    const int mrow = lane & 15;     // row within tile for A, col for B
    const int kofs = (lane >> 4) * 2;

    const float* xr = x + ((size_t)b * TLEN + (i0 + mrow)) * CDIM + kofs;
    const float* yr = y + ((size_t)b * TLEN + (j0 + mrow)) * CDIM + kofs;

    v8f acc = {};
#pragma unroll
    for (int kk = 0; kk < CDIM; kk += 4) {
        const float2 av = *(const float2*)(xr + kk);
        const float2 bv = *(const float2*)(yr + kk);
        v2f A; A.x = av.x; A.y = av.y;
        v2f B; B.x = bv.x; B.y = bv.y;
        // 8 args: (neg_a, A, neg_b, B, c_mod, C, reuse_a, reuse_b)
        acc = __builtin_amdgcn_wmma_f32_16x16x4_f32(
            false, A, false, B, (short)0, acc, false, false);
    }

    // Write back D-tile: element (i0 + r + 8*hi, j0 + (lane&15))
    float* outp = xy + (size_t)b * TLEN * TLEN;
    const int n  = j0 + (lane & 15);
    const int hi = lane >> 4;
#pragma unroll
    for (int r = 0; r < 8; ++r) {
        outp[(size_t)(i0 + r + 8 * hi) * TLEN + n] = acc[r];
    }
}

// ---------------------------------------------------------------------------
// Kernel 2: SoftDTW DP over anti-diagonals, one workgroup per batch.
// 512 threads (16 wave32). Thread tid owns DP row i = tid+1.
// Cost(i,j) = x2[i-1] + y2[j-1] - 2*XY[i-1,j-1]; each thread streams its
// own XY row sequentially (one new float per diagonal step).
// softmin (gamma=1): mn - log(exp(mn-a)+exp(mn-u)+exp(mn-l))
// ---------------------------------------------------------------------------
__global__ __launch_bounds__(512) void
softdtw_dp_kernel(const float* __restrict__ x,
                  const float* __restrict__ y,
                  const float* __restrict__ xy,
                  float* __restrict__ out)
{
    const int b   = blockIdx.x;
    const int tid = threadIdx.x;          // 0..511

    __shared__ float y2s[TLEN];
    __shared__ float buf[3][TLEN + 1];

    // Row norms: x2 kept in a register (this thread's row), y2 shared in LDS.
    const float* xr = x + ((size_t)b * TLEN + tid) * CDIM;
    const float* yr = y + ((size_t)b * TLEN + tid) * CDIM;
    float x2 = 0.0f, y2 = 0.0f;
#pragma unroll
    for (int c = 0; c < CDIM; c += 4) {
        const float4 xv = *(const float4*)(xr + c);
        const float4 yv = *(const float4*)(yr + c);
        x2 += xv.x * xv.x + xv.y * xv.y + xv.z * xv.z + xv.w * xv.w;
        y2 += yv.x * yv.x + yv.y * yv.y + yv.z * yv.z + yv.w * yv.w;
    }
    y2s[tid] = y2;

    // Diagonal init: buf[0] = d0 (R[0,0]=0, rest INF), buf[1] = d1 (all INF)
    buf[0][tid + 1] = INF_V;
    buf[1][tid + 1] = INF_V;
    if (tid == 0) {
        buf[0][0] = 0.0f;
        buf[1][0] = INF_V;
    }
    __syncthreads();

    const int   i    = tid + 1;                 // DP row index, 1..512
    const float* row = xy + (size_t)b * TLEN * TLEN + (size_t)tid * TLEN;

    for (int k = 2; k <= 2 * TLEN; ++k) {
        const int s = k - 2;
        const float* prev2 = buf[s % 3];
        const float* prev  = buf[(s + 1) % 3];
        float*       cur   = buf[(s + 2) % 3];

        const int j = k - i;
        float res = INF_V;
        if (j >= 1 && j <= TLEN) {
            const float a  = prev2[i - 1];      // diag  R[i-1,j-1]
            const float u  = prev[i - 1];       // up    R[i-1,j]
            const float l  = prev[i];           // left  R[i,j-1]
            float mn = fminf(a, fminf(u, l));
            const float sm = mn - logf(expf(mn - a) + expf(mn - u) + expf(mn - l));
            const float cost = x2 + y2s[j - 1] - 2.0f * row[j - 1];
            res = cost + sm;
        }
        cur[i] = res;
        if (tid == 0) cur[0] = INF_V;
        __syncthreads();
    }

    // Final diagonal is buf[(2 + (2*TLEN - 2)) % 3]; answer at entry i = N.
    if (tid == 0) {
        out[b] = buf[(2 * TLEN) % 3][TLEN];
    }
}

// ---------------------------------------------------------------------------
// Launch: d_in[0]=x (128*512*64 f32), d_in[1]=y, d_out = 128 f32,
// d_ws holds XY (128*512*512 f32 = 134,217,728 bytes).
// ---------------------------------------------------------------------------
extern "C" void kernel_launch(void* const* d_in, const int* in_sizes, int n_in,
                              void* d_out, int out_size, void* d_ws, size_t ws_size,
                              hipStream_t stream) {
    const float* x = (const float*)d_in[0];
    const float* y = (const float*)d_in[1];
    float* xy  = (float*)d_ws;
    float* out = (float*)d_out;

    // 131072 tiles total, 8 waves (tiles) per 256-thread block -> 16384 blocks
    const int tiles  = BATCH * (TLEN / 16) * (TLEN / 16);
    const int blocks = tiles / 8;
    softdtw_xy_wmma_kernel<<<blocks, 256, 0, stream>>>(x, y, xy);

    softdtw_dp_kernel<<<BATCH, TLEN, 0, stream>>>(x, y, xy, out);
}